// HausdorffErosion3d_30846455119933
// MI455X (gfx1250) — compile-verified
//
#include <hip/hip_runtime.h>

#define GLOBAL_AS __attribute__((address_space(1)))
#define LDS_AS    __attribute__((address_space(3)))

typedef __attribute__((ext_vector_type(2))) float v2f;
typedef __attribute__((ext_vector_type(4))) float v4f;
typedef __attribute__((ext_vector_type(8))) float v8f;
typedef __attribute__((ext_vector_type(4))) int   v4i;

#define VOL   2097152   // 128^3
#define HW    16384     // 128*128
#define ROWW  136       // padded LDS row: [0..3] pad (data starts at 4 -> 16B aligned), [132..135] pad
#define NROW  18        // 16 interior rows + 2 halo
#define PLANE (NROW * ROWW)   // 2448 floats per plane

// ---------------- CDNA5 async global->LDS staging ----------------
__device__ __forceinline__ void async_load_b128(const float* g, float* l) {
#if __has_builtin(__builtin_amdgcn_global_load_async_to_lds_b128)
  __builtin_amdgcn_global_load_async_to_lds_b128(
      (GLOBAL_AS v4i*)g, (LDS_AS v4i*)l, /*imm offset*/0, /*cpol*/0);
#else
  unsigned loff = (unsigned)(unsigned long long)(LDS_AS void*)l;
  asm volatile("global_load_async_to_lds_b128 %0, %1, off"
               :: "v"(loff), "v"(g) : "memory");
#endif
}

__device__ __forceinline__ void wait_async0() {
#if __has_builtin(__builtin_amdgcn_s_wait_asynccnt)
  __builtin_amdgcn_s_wait_asynccnt(0);
#else
  asm volatile("s_wait_asynccnt 0" ::: "memory");
#endif
}

// ---------------- WMMA wave reduction ----------------
// A = ones(16x4) -> D[i][j] = sum_k B[k][j]. Every lane's partial occupies
// exactly one B slot, so summing one D row's 16 columns yields the wave total.
// d[0] lanes 0-15 hold colsums (row M=0); lanes 16-31 hold the identical row M=8.
__device__ __forceinline__ float wave_reduce_wmma(float p) {
  v2f a; a.x = 1.0f; a.y = 1.0f;   // A = all ones
  v2f b; b.x = p;    b.y = 0.0f;   // one contribution per lane
  v8f c = {};
  c = __builtin_amdgcn_wmma_f32_16x16x4_f32(false, a, false, b, (short)0, c,
                                            false, false);
  float s = c[0];
  s += __shfl_xor(s, 1);
  s += __shfl_xor(s, 2);
  s += __shfl_xor(s, 4);
  s += __shfl_xor(s, 8);   // every lane now holds the full wave sum
  return s;
}

// ---------------- Kernel 1: softmax + one-hot diff squared ----------------
__global__ __launch_bounds__(256) void prep_kernel(const float* __restrict__ in,
                                                   const int* __restrict__ tgt,
                                                   float* __restrict__ op) {
  size_t i = (size_t)blockIdx.x * 256 + threadIdx.x;   // site in B*D*H*W = 4194304
  size_t bb = i >> 21;           // batch
  size_t r  = i & 2097151;       // d*HW + h*W + w
  const float* x = in + bb * 4 * (size_t)VOL + r;
  float x0 = x[0], x1 = x[VOL], x2 = x[2 * (size_t)VOL], x3 = x[3 * (size_t)VOL];
  float m = fmaxf(fmaxf(x0, x1), fmaxf(x2, x3));
  float e0 = __expf(x0 - m), e1 = __expf(x1 - m);
  float e2 = __expf(x2 - m), e3 = __expf(x3 - m);
  float inv = 1.0f / (e0 + e1 + e2 + e3);
  int t = tgt[i];
  float d0 = e0 * inv - (t == 0 ? 1.0f : 0.0f);
  float d1 = e1 * inv - (t == 1 ? 1.0f : 0.0f);
  float d2 = e2 * inv - (t == 2 ? 1.0f : 0.0f);
  float d3 = e3 * inv - (t == 3 ? 1.0f : 0.0f);
  float* o = op + bb * 4 * (size_t)VOL + r;
  o[0] = d0 * d0;
  o[VOL] = d1 * d1;
  o[2 * (size_t)VOL] = d2 * d2;
  o[3 * (size_t)VOL] = d3 * d3;
}

// ---------------- Kernel 2: one erosion step + partial reduction ----------------
__device__ __forceinline__ void load_plane(float* lds, int slot,
                                           const float* __restrict__ srcb,
                                           int d, int h0) {
  const int tid = threadIdx.x;
  const bool dvalid = (0 <= d) & (d < 128);
  // 18 rows x 32 b128 chunks = 576 lane-tasks
  for (int t = tid; t < NROW * 32; t += 256) {
    int rr = t >> 5;            // 0..17
    int cw = (t & 31) << 2;     // 0,4,...,124
    int gh = h0 - 1 + rr;
    float* dst = lds + slot * PLANE + rr * ROWW + 4 + cw;
    if (dvalid && (unsigned)gh < 128u) {
      const float* g = srcb + (size_t)d * HW + (size_t)gh * 128 + cw;
      async_load_b128(g, dst);
    } else {
      v4f z = {};
      *(v4f*)dst = z;           // ds_store_b128 of zeros (halo row / halo plane)
    }
  }
}

__global__ __launch_bounds__(256) void erode_kernel(const float* __restrict__ src,
                                                    float* __restrict__ dst,
                                                    const float* __restrict__ kern,
                                                    const float* __restrict__ bias,
                                                    const float* __restrict__ weight,
                                                    float* __restrict__ partials,
                                                    float coeff) {
  __shared__ float lds[4 * PLANE];
  __shared__ float sred[8];

  const int z  = blockIdx.z;            // b*4 + c, 0..7
  const int h0 = blockIdx.x * 16;
  const int d0 = blockIdx.y * 16;
  const size_t base = (size_t)z * VOL;
  const float* srcb = src + base;
  float* dstb = dst + base;

  const float scale = kern[(z & 3) * 27 + 13];   // center tap of the cross kernel
  const float biasv = bias[z & 3];
  const float wc = weight[z] * coeff;

  // zero the left/right pad columns once; async loads never touch them
  for (int t = threadIdx.x; t < 4 * NROW; t += 256) {
    float* row = lds + (t / NROW) * PLANE + (t % NROW) * ROWW;
    row[3] = 0.0f;
    row[132] = 0.0f;
  }

  // preload planes d0-1, d0, d0+1 into slots (d+1)&3
  load_plane(lds, (d0    ) & 3, srcb, d0 - 1, h0);
  load_plane(lds, (d0 + 1) & 3, srcb, d0,     h0);
  load_plane(lds, (d0 + 2) & 3, srcb, d0 + 1, h0);
  wait_async0();
  __syncthreads();

  const int w = threadIdx.x & 127;
  float p = 0.0f;

  for (int d = d0; d < d0 + 16; ++d) {
    // prefetch plane d+2 (needed at iteration d+1); slot (d+3)&3 == slot of d-2
    if (d + 2 <= d0 + 16) load_plane(lds, (d + 3) & 3, srcb, d + 2, h0);

    const float* pm = lds + ((d    ) & 3) * PLANE;   // plane d-1
    const float* pc = lds + ((d + 1) & 3) * PLANE;   // plane d
    const float* pp = lds + ((d + 2) & 3) * PLANE;   // plane d+1

    for (int hh = (threadIdx.x >> 7); hh < 16; hh += 2) {
      int rc = (hh + 1) * ROWW + 4 + w;
      float s = pc[rc] + pc[rc - 1] + pc[rc + 1]
              + pc[rc - ROWW] + pc[rc + ROWW]
              + pm[rc] + pp[rc];
      float y = fmaxf(fmaf(s, scale, biasv), 0.0f);
      dstb[(size_t)d * HW + (size_t)(h0 + hh) * 128 + w] = y;
      p += y;
    }
    wait_async0();
    __syncthreads();
  }

  // block reduction: WMMA per-wave, then 8 wave sums via LDS (deterministic)
  float wsum = wave_reduce_wmma(p);
  if ((threadIdx.x & 31) == 0) sred[threadIdx.x >> 5] = wsum;
  __syncthreads();
  if (threadIdx.x == 0) {
    float t = 0.0f;
    for (int k = 0; k < 8; ++k) t += sred[k];
    int idx = (blockIdx.z * gridDim.y + blockIdx.y) * gridDim.x + blockIdx.x;
    partials[idx] = wc * t;
  }
}

// ---------------- Kernel 3: deterministic final reduction ----------------
__global__ __launch_bounds__(256) void final_reduce(const float* __restrict__ partials,
                                                    int n, float* __restrict__ out,
                                                    float invN) {
  __shared__ float sred[8];
  float p = 0.0f;
  for (int i = threadIdx.x; i < n; i += 256) p += partials[i];
  float wsum = wave_reduce_wmma(p);
  if ((threadIdx.x & 31) == 0) sred[threadIdx.x >> 5] = wsum;
  __syncthreads();
  if (threadIdx.x == 0) {
    float t = 0.0f;
    for (int k = 0; k < 8; ++k) t += sred[k];
    out[0] = t * invN;
  }
}

// ---------------- Host launcher ----------------
extern "C" void kernel_launch(void* const* d_in, const int* in_sizes, int n_in,
                              void* d_out, int out_size, void* d_ws, size_t ws_size,
                              hipStream_t stream) {
  const float* inp  = (const float*)d_in[0];
  const int*   tgt  = (const int*)d_in[1];
  const float* wgt  = (const float*)d_in[2];
  const float* kern = (const float*)d_in[3];
  const float* bias = (const float*)d_in[4];

  float* bufA = (float*)d_ws;                 // 16777216 floats
  float* bufB = bufA + (size_t)8 * VOL;       // 16777216 floats
  float* partials = bufB + (size_t)8 * VOL;   // 10 * 512 floats

  // op0 = (softmax(input) - onehot(target))^2
  prep_kernel<<<16384, 256, 0, stream>>>(inp, tgt, bufA);

  dim3 grid(8, 8, 8);   // h-tiles x d-chunks x (b*c)
  for (int k = 0; k < 10; ++k) {
    float coeff = (float)((k + 1) * (k + 1));
    const float* s = (k & 1) ? bufB : bufA;
    float*       d = (k & 1) ? bufA : bufB;
    erode_kernel<<<grid, 256, 0, stream>>>(s, d, kern, bias, wgt,
                                           partials + k * 512, coeff);
  }

  final_reduce<<<1, 256, 0, stream>>>(partials, 10 * 512, (float*)d_out,
                                      1.0f / 16777216.0f);
}